// E2RFuncttion_75041668596272
// MI455X (gfx1250) — compile-verified
//
#include <hip/hip_runtime.h>
#include <hip/hip_bf16.h>

// ---- problem constants (match reference) ----
#define IN_DIM   128
#define CAT      256          // 2*IN_DIM
#define HID      256
#define OUTD     128
#define N_NODES  100000
#define N_EDGES  500000
#define N_MC     2
#define TILES_PER_MC (N_EDGES / 16)        // 31250
#define N_TILES      (N_MC * TILES_PER_MC) // 62500
#define WAVES_PER_BLK 4
#define FSTRIDE  132   // feat row stride (floats): 128 data + 4 TDM pad -> banks (4m+k)%64 unique
#define HSTRIDE  260   // hid  row stride (floats): banks (4m+k)%64 unique

typedef float v2f __attribute__((ext_vector_type(2)));
typedef float v8f __attribute__((ext_vector_type(8)));
typedef unsigned int u32x4 __attribute__((ext_vector_type(4)));
typedef int i32x8 __attribute__((ext_vector_type(8)));
typedef int i32x4 __attribute__((ext_vector_type(4)));

// TDM gather: DMA 8 rows (each 128 f32) of the node table into LDS.
// D# bitfields per CDNA5 ISA (gather mode, 32-bit indices,
// data_size=4B, pad_enable: 4 dwords of pad per 128-dword row -> 132-dword stride).
// This toolchain's builtin is the 6-arg clang-23 form:
//   (u32x4 g0, i32x8 g1, i32x4 g2, i32x4 g3, i32x8 extra/unused, i32 cpol)
__device__ __forceinline__ void tdm_gather8(unsigned ldsByteOff, const float* base,
                                            const unsigned* idx /* 8 SGPR row ids */) {
    const unsigned long long ga = (unsigned long long)(uintptr_t)base;
    u32x4 g0;
    g0[0] = 1u            // count = 1 (valid descriptor)
          | (1u << 30)    // gather_index_size = 1 (32-bit indices)
          | (1u << 31);   // gather_mode = 1
    g0[1] = ldsByteOff;                                   // lds_addr (bytes)
    g0[2] = (unsigned)ga;                                 // global_addr[31:0]
    g0[3] = (unsigned)((ga >> 32) & 0x01FFFFFFull)        // global_addr[56:32]
          | (2u << 30);                                   // type = 2 ("image")
    i32x8 g1;
    g1[0] = (int)((2u << 16)    // data_size = 2 -> 4 bytes
                | (1u << 20)    // pad_enable
                | (6u << 22)    // pad_interval = 6 -> every 128 dwords
                | (3u << 25));  // pad_amount = 3 -> 4 dwords
    g1[1] = (int)(128u << 16);                              // tensor_dim0 = 128 (low16 @ bits 63:48)
    g1[2] = (int)(((unsigned)N_NODES & 0xFFFFu) << 16);     // tensor_dim1 low16 @ bits 95:80
    g1[3] = (int)((((unsigned)N_NODES >> 16) & 0xFFFFu)     // tensor_dim1 high16
                | (128u << 16));                            // tile_dim0 = 128
    g1[4] = 8;      // tile_dim1 = 8 valid indices
    g1[5] = 128;    // tensor_dim0_stride = 128 elements
    g1[6] = 0;
    g1[7] = 0;
    i32x4 g2, g3;
    g2[0] = (int)idx[0]; g2[1] = (int)idx[1]; g2[2] = (int)idx[2]; g2[3] = (int)idx[3];
    g3[0] = (int)idx[4]; g3[1] = (int)idx[5]; g3[2] = (int)idx[6]; g3[3] = (int)idx[7];
    i32x8 gx = { 0, 0, 0, 0, 0, 0, 0, 0 };                  // unused group (VADDR4 / NULL)
    __builtin_amdgcn_tensor_load_to_lds(g0, g1, g2, g3, gx, 0);
}

__global__ __launch_bounds__(32 * WAVES_PER_BLK)
void edge_mlp_wmma_f32(const long long* __restrict__ eidx,   // (2, N_EDGES) int64
                       const float*     __restrict__ x,      // (N_MC, N_NODES, IN_DIM)
                       const float*     __restrict__ W1,     // (256, 256) row-major
                       const float*     __restrict__ b1,     // (256)
                       const float*     __restrict__ W2,     // (256, 128) row-major
                       const float*     __restrict__ b2,     // (128)
                       float*           __restrict__ out)    // (N_MC, N_EDGES, OUTD)
{
    __shared__ float lds_featS[WAVES_PER_BLK][16 * FSTRIDE]; // x[src] rows (TDM gather dest)
    __shared__ float lds_featD[WAVES_PER_BLK][16 * FSTRIDE]; // x[dst] rows (TDM gather dest)
    __shared__ float lds_hid  [WAVES_PER_BLK][16 * HSTRIDE]; // hidden tile (post-ReLU)

    const int lane = threadIdx.x & 31;
    const int wave = threadIdx.x >> 5;
    const int tile = blockIdx.x * WAVES_PER_BLK + wave;       // grid sized exactly: no guard
    const int mc   = tile / TILES_PER_MC;
    const int t    = tile - mc * TILES_PER_MC;
    const long long e0 = (long long)t * 16;

    const float* xm = x + (size_t)mc * N_NODES * IN_DIM;
    float* featS = lds_featS[wave];
    float* featD = lds_featD[wave];
    float* hid   = lds_hid[wave];

    const int lrow  = lane & 15;            // M (or N) index within 16
    const int khalf = (lane >> 4) << 1;     // 0 for lanes 0-15, 2 for lanes 16-31

    // ------------- gather 16 edges via Tensor Data Mover (4 descriptors) ----
    {
        // node ids: wave-uniform loads -> SGPRs via readfirstlane (32-bit ids)
        const unsigned* e32 = (const unsigned*)eidx;   // int64 little-endian, ids < 2^31
        unsigned s_id[16], d_id[16];
        #pragma unroll
        for (int m = 0; m < 16; ++m) {
            s_id[m] = (unsigned)__builtin_amdgcn_readfirstlane((int)e32[2 * (e0 + m)]);
            d_id[m] = (unsigned)__builtin_amdgcn_readfirstlane((int)e32[2 * ((long long)N_EDGES + e0 + m)]);
        }
        const unsigned sOff = (unsigned)(uintptr_t)featS;
        const unsigned dOff = (unsigned)(uintptr_t)featD;
        __asm__ volatile("" ::: "memory");
        tdm_gather8(sOff,                          xm, &s_id[0]);   // src rows 0..7
        tdm_gather8(sOff + 8u * FSTRIDE * 4u,      xm, &s_id[8]);   // src rows 8..15
        tdm_gather8(dOff,                          xm, &d_id[0]);   // dst rows 0..7
        tdm_gather8(dOff + 8u * FSTRIDE * 4u,      xm, &d_id[8]);   // dst rows 8..15
        __builtin_amdgcn_s_wait_tensorcnt((short)0);
        __asm__ volatile("" ::: "memory");
    }

    // ---------------- GEMM1: h = [xs|xd](16x256) @ W1(256x256) + b1, ReLU ---
    for (int j = 0; j < 16; ++j) {                   // hidden n-block (16 cols)
        const float bv = b1[j * 16 + lrow];
        v8f acc = { bv, bv, bv, bv, bv, bv, bv, bv };
        const int colB = j * 16 + lrow;

        #pragma unroll 4
        for (int kk = 0; kk < IN_DIM; kk += 4) {     // K = 0..127  (src half, W1a)
            v2f a = *(const v2f*)(featS + lrow * FSTRIDE + kk + khalf);
            v2f b;
            b.x = W1[(size_t)(kk + khalf)     * HID + colB];
            b.y = W1[(size_t)(kk + khalf + 1) * HID + colB];
            acc = __builtin_amdgcn_wmma_f32_16x16x4_f32(false, a, false, b,
                                                        (short)0, acc, false, false);
        }
        #pragma unroll 4
        for (int kk = 0; kk < IN_DIM; kk += 4) {     // K = 128..255 (dst half, W1b)
            v2f a = *(const v2f*)(featD + lrow * FSTRIDE + kk + khalf);
            v2f b;
            b.x = W1[(size_t)(IN_DIM + kk + khalf)     * HID + colB];
            b.y = W1[(size_t)(IN_DIM + kk + khalf + 1) * HID + colB];
            acc = __builtin_amdgcn_wmma_f32_16x16x4_f32(false, a, false, b,
                                                        (short)0, acc, false, false);
        }
        // ReLU + spill hidden n-block to LDS in C/D layout:
        // vgpr r holds row (r + 8*(lane>=16)), col = lane&15
        #pragma unroll
        for (int r = 0; r < 8; ++r) {
            const float v = acc[r] > 0.f ? acc[r] : 0.f;
            hid[(r + 8 * (lane >> 4)) * HSTRIDE + j * 16 + lrow] = v;
        }
    }
    __asm__ volatile("s_wait_dscnt 0" ::: "memory");

    // ---------------- GEMM2: out = hid(16x256) @ W2(256x128) + b2 ------------
    float* outp = out + ((size_t)mc * N_EDGES + (size_t)e0) * OUTD;
    for (int n = 0; n < 8; ++n) {                    // output n-block (16 cols)
        const float bv = b2[n * 16 + lrow];
        v8f acc = { bv, bv, bv, bv, bv, bv, bv, bv };
        const int colB = n * 16 + lrow;

        #pragma unroll 4
        for (int kk = 0; kk < HID; kk += 4) {
            v2f a = *(const v2f*)(hid + lrow * HSTRIDE + kk + khalf);
            v2f b;
            b.x = W2[(size_t)(kk + khalf)     * OUTD + colB];
            b.y = W2[(size_t)(kk + khalf + 1) * OUTD + colB];
            acc = __builtin_amdgcn_wmma_f32_16x16x4_f32(false, a, false, b,
                                                        (short)0, acc, false, false);
        }
        // store C/D layout directly: lanes 0-15 write rows r, lanes 16-31 rows r+8
        #pragma unroll
        for (int r = 0; r < 8; ++r) {
            outp[(size_t)(r + 8 * (lane >> 4)) * OUTD + n * 16 + lrow] = acc[r];
        }
    }
}

extern "C" void kernel_launch(void* const* d_in, const int* in_sizes, int n_in,
                              void* d_out, int out_size, void* d_ws, size_t ws_size,
                              hipStream_t stream) {
    (void)in_sizes; (void)n_in; (void)d_ws; (void)ws_size; (void)out_size;
    const long long* eidx = (const long long*)d_in[0];   // int64 edge_index (2, N_EDGES)
    const float*     x    = (const float*)    d_in[1];   // (N_MC, N_NODES, IN_DIM)
    const float*     W1   = (const float*)    d_in[2];   // (256, 256)
    const float*     b1   = (const float*)    d_in[3];   // (256)
    const float*     W2   = (const float*)    d_in[4];   // (256, 128)
    const float*     b2   = (const float*)    d_in[5];   // (128)
    float*           out  = (float*)          d_out;     // (N_MC, N_EDGES, OUTD)

    const int nblocks = N_TILES / WAVES_PER_BLK;         // 62500 / 4 = 15625 exact
    edge_mlp_wmma_f32<<<nblocks, 32 * WAVES_PER_BLK, 0, stream>>>(
        eidx, x, W1, b1, W2, b2, out);
}